// GSN_8392366096421
// MI455X (gfx1250) — compile-verified
//
#include <hip/hip_runtime.h>
#include <math.h>

// Problem constants: N_NODES=3, T=2048, D=2048  ->  T*D = 2^22
#define LOG_TD 22u
#define TD (1u << LOG_TD)            // 4194304 elements per (T,D) slab

typedef __attribute__((ext_vector_type(2))) float v2f;
typedef __attribute__((ext_vector_type(8))) float v8f;

constexpr unsigned TPW = 4u;         // WMMA tiles per wave

// ---- fast transcendentals on hardware trans units ------------------------
__device__ __forceinline__ float hw_exp2(float x) {
#if __has_builtin(__builtin_amdgcn_exp2f)
  return __builtin_amdgcn_exp2f(x);
#else
  return exp2f(x);
#endif
}
__device__ __forceinline__ float hw_rcp(float x) {
#if __has_builtin(__builtin_amdgcn_rcpf)
  return __builtin_amdgcn_rcpf(x);
#else
  return 1.0f / x;
#endif
}
__device__ __forceinline__ float fast_sigmoid(float x) {
  // 1/(1+e^-x); large |x| saturates correctly through inf/0 of hw exp2/rcp
  return hw_rcp(1.0f + hw_exp2(-1.4426950408889634f * x));
}
__device__ __forceinline__ float fast_tanh(float x) {
#if __has_builtin(__builtin_amdgcn_tanhf)
  return __builtin_amdgcn_tanhf(x);     // gfx1250 v_tanh_f32
#else
  // tanh = 1 - 2/(e^{2x}+1); x>>0 -> exp2=inf -> rcp=0 -> 1;  x<<0 -> -1
  return 1.0f - 2.0f * hw_rcp(hw_exp2(2.8853900817779268f * x) + 1.0f);
#endif
}

// A_flat for the gate linear: [msg_full (3*TD) ; h broadcast (3*TD)]
template <bool HAS_M1>
__device__ __forceinline__ float gate_src(unsigned m,
    const float* __restrict__ hv, const float* __restrict__ m0,
    const float* __restrict__ m1) {
  if (m < 3u * TD) {                       // mailbox region
    unsigned n = m >> LOG_TD;              // slot 0..2
    unsigned r = m & (TD - 1u);
    if (n == 0u) return m0[r];
    if (HAS_M1 && n == 1u) return m1[r];
    return 0.0f;                           // empty mailbox slots are zero
  }
  unsigned r = (m - 3u * TD) & (TD - 1u);  // h broadcast region
  return hv[r];
}

// B_flat for the cand linear: per slot n: [gate[n]*msg[n] (TD) ; h (TD)]
template <bool HAS_M1>
__device__ __forceinline__ float cand_src(unsigned m,
    const float* __restrict__ hv, const float* __restrict__ m0,
    const float* __restrict__ m1, const float* __restrict__ G) {
  unsigned seg = m >> LOG_TD;              // 0..5
  unsigned r   = m & (TD - 1u);
  if (seg & 1u) return hv[r];
  unsigned n = seg >> 1;
  if (n == 0u) return G[r] * m0[r];
  if (HAS_M1 && n == 1u) return G[TD + r] * m1[r];
  return 0.0f;
}

// ---------------- Pass 1: gate  G[3*TD] = sigmoid((TD,6)@(6,3)+b) ----------
template <bool HAS_M1>
__global__ __launch_bounds__(256) void gsn_gate_wmma(
    const float* __restrict__ hv, const float* __restrict__ m0,
    const float* __restrict__ m1, const float* __restrict__ Wx,  // (3,6)
    const float* __restrict__ Bx,                                // (3,)
    float* __restrict__ G) {
  const unsigned lane  = threadIdx.x & 31u;
  const unsigned wave  = threadIdx.x >> 5u;
  const unsigned tile0 = (blockIdx.x * 8u + wave) * TPW;

  const bool     lo  = lane < 16u;
  const unsigned N   = lane & 15u;         // output column (j) for this lane
  const bool     nok = N < 3u;
  const unsigned wn  = nok ? N : 0u;       // clamped safe index
  const float    msk = nok ? 1.0f : 0.0f;
  const float w0 = Wx[wn*6u+0u]*msk, w1 = Wx[wn*6u+1u]*msk, w2 = Wx[wn*6u+2u]*msk,
              w3 = Wx[wn*6u+3u]*msk, w4 = Wx[wn*6u+4u]*msk, w5 = Wx[wn*6u+5u]*msk;
  const float bias = Bx[wn] * msk;

  // B 4x16 (f32): lane L<16 -> (N=L, K={0,1}); L>=16 -> (N=L-16, K={2,3})
  v2f b, b2;
  b.x  = lo ? w0 : w2;
  b.y  = lo ? w1 : w3;
  b2.x = lo ? w4 : 0.0f;                   // K=4 | K=6(pad)->0
  b2.y = lo ? w5 : 0.0f;                   // K=5 | K=7(pad)->0

  for (unsigned t = 0; t < TPW; ++t) {
    const unsigned q0   = (tile0 + t) * 16u;   // first chunk (A-row) of tile
    const unsigned row  = q0 + (lane & 15u);
    const unsigned base = 6u * row;
    // A tile 16x4 (f32): lane L<16 -> (M=L, K={0,1}); L>=16 -> (M, K={2,3})
    v2f a, a2;
    a.x  = gate_src<HAS_M1>(base + (lo ? 0u : 2u), hv, m0, m1);
    a.y  = gate_src<HAS_M1>(base + (lo ? 1u : 3u), hv, m0, m1);
    a2.x = lo ? gate_src<HAS_M1>(base + 4u, hv, m0, m1) : 0.0f;
    a2.y = lo ? gate_src<HAS_M1>(base + 5u, hv, m0, m1) : 0.0f;

    v8f c;
#pragma unroll
    for (int v = 0; v < 8; ++v) c[v] = bias;

    v8f d = __builtin_amdgcn_wmma_f32_16x16x4_f32(false, a,  false, b,  (short)0, c, false, false);
    d     = __builtin_amdgcn_wmma_f32_16x16x4_f32(false, a2, false, b2, (short)0, d, false, false);

    if (nok) {                             // D: VGPR v -> M = v + (lo?0:8)
      const unsigned mb = lo ? 0u : 8u;
#pragma unroll
      for (int v = 0; v < 8; ++v) {
        unsigned p = 3u * (q0 + mb + (unsigned)v) + N;
        G[p] = fast_sigmoid(d[v]);
      }
    }
  }
}

// ------------- Pass 2: cand + fused delta contribution into CT -------------
template <bool HAS_M1>
__global__ __launch_bounds__(256) void gsn_cand_wmma(
    const float* __restrict__ hv, const float* __restrict__ m0,
    const float* __restrict__ m1, const float* __restrict__ Ww,  // (3,6)
    const float* __restrict__ Bw,                                // (3,)
    const float* __restrict__ G, float* __restrict__ CT) {
  const unsigned lane  = threadIdx.x & 31u;
  const unsigned wave  = threadIdx.x >> 5u;
  const unsigned tile0 = (blockIdx.x * 8u + wave) * TPW;

  const bool     lo  = lane < 16u;
  const unsigned N   = lane & 15u;
  const bool     nok = N < 3u;
  const unsigned wn  = nok ? N : 0u;
  const float    msk = nok ? 1.0f : 0.0f;
  const float w0 = Ww[wn*6u+0u]*msk, w1 = Ww[wn*6u+1u]*msk, w2 = Ww[wn*6u+2u]*msk,
              w3 = Ww[wn*6u+3u]*msk, w4 = Ww[wn*6u+4u]*msk, w5 = Ww[wn*6u+5u]*msk;
  const float bias = Bw[wn] * msk;

  v2f b, b2;
  b.x  = lo ? w0 : w2;
  b.y  = lo ? w1 : w3;
  b2.x = lo ? w4 : 0.0f;
  b2.y = lo ? w5 : 0.0f;

  for (unsigned t = 0; t < TPW; ++t) {
    const unsigned q0   = (tile0 + t) * 16u;
    const unsigned row  = q0 + (lane & 15u);
    const unsigned base = 6u * row;
    v2f a, a2;
    a.x  = cand_src<HAS_M1>(base + (lo ? 0u : 2u), hv, m0, m1, G);
    a.y  = cand_src<HAS_M1>(base + (lo ? 1u : 3u), hv, m0, m1, G);
    a2.x = lo ? cand_src<HAS_M1>(base + 4u, hv, m0, m1, G) : 0.0f;
    a2.y = lo ? cand_src<HAS_M1>(base + 5u, hv, m0, m1, G) : 0.0f;

    v8f c;
#pragma unroll
    for (int v = 0; v < 8; ++v) c[v] = bias;

    v8f d = __builtin_amdgcn_wmma_f32_16x16x4_f32(false, a,  false, b,  (short)0, c, false, false);
    d     = __builtin_amdgcn_wmma_f32_16x16x4_f32(false, a2, false, b2, (short)0, d, false, false);

    if (nok) {
      const unsigned mb = lo ? 0u : 8u;
#pragma unroll
      for (int v = 0; v < 8; ++v) {
        unsigned p   = 3u * (q0 + mb + (unsigned)v) + N;
        float cand   = fast_tanh(d[v]);
        float g      = G[p];
        unsigned n_o = p >> LOG_TD;        // slot of this (n,t,d) position
        unsigned r   = p & (TD - 1u);
        float mv;
        if (n_o == 0u)               mv = m0[r];
        else if (HAS_M1 && n_o == 1u) mv = m1[r];
        else                          mv = 0.0f;
        CT[p] = (1.0f - g) * mv + g * cand;   // per-slot delta contribution
      }
    }
  }
}

// ---------------- Pass 3: out = h + sum_n CT[n*TD + r] ---------------------
__global__ __launch_bounds__(256) void gsn_finalize(
    const float* __restrict__ hv, const float* __restrict__ CT,
    float* __restrict__ out) {
  const unsigned i = (blockIdx.x * 256u + threadIdx.x) * 4u;
  const float4 h4 = *(const float4*)(hv + i);
  const float4 c0 = *(const float4*)(CT + i);
  const float4 c1 = *(const float4*)(CT + TD + i);
  const float4 c2 = *(const float4*)(CT + 2u * TD + i);
  float4 o;
  o.x = h4.x + c0.x + c1.x + c2.x;
  o.y = h4.y + c0.y + c1.y + c2.y;
  o.z = h4.z + c0.z + c1.z + c2.z;
  o.w = h4.w + c0.w + c1.w + c2.w;
  *(float4*)(out + i) = o;
}

extern "C" void kernel_launch(void* const* d_in, const int* in_sizes, int n_in,
                              void* d_out, int out_size, void* d_ws, size_t ws_size,
                              hipStream_t stream) {
  (void)in_sizes; (void)n_in; (void)out_size; (void)ws_size;
  const float* feat = (const float*)d_in[0];   // (3, T, D)
  const float* W_w  = (const float*)d_in[1];   // (3, 6)  -- cand linear
  const float* W_b  = (const float*)d_in[2];   // (3,)
  const float* Wx_w = (const float*)d_in[3];   // (3, 6)  -- gate linear
  const float* Wx_b = (const float*)d_in[4];   // (3,)
  float* out = (float*)d_out;

  float* G  = (float*)d_ws;                    // 3*TD floats (gate, flat)
  float* CT = G + (size_t)3 * TD;              // 3*TD floats (delta contribs)

  const float* h0 = feat;
  const float* h1 = feat + (size_t)TD;
  const float* h2 = feat + (size_t)2 * TD;

  // Graph: node0 <- [h2]; node1 <- [h0]; node2 <- [h1, h0] (mailbox order)
  dim3 blk(256);
  dim3 grd_mm(TD / 16u / TPW / 8u);  // 8192 blocks: 8 waves x 4 tiles each
  dim3 grd_fin(TD / 4u / 256u);      // float4 elementwise -> 4096 blocks

  // ---- node 0: mailbox [h2] (deg 1) ----
  gsn_gate_wmma<false><<<grd_mm, blk, 0, stream>>>(h0, h2, nullptr, Wx_w, Wx_b, G);
  gsn_cand_wmma<false><<<grd_mm, blk, 0, stream>>>(h0, h2, nullptr, W_w, W_b, G, CT);
  gsn_finalize<<<grd_fin, blk, 0, stream>>>(h0, CT, out + 0 * (size_t)TD);

  // ---- node 1: mailbox [h0] (deg 1) ----
  gsn_gate_wmma<false><<<grd_mm, blk, 0, stream>>>(h1, h0, nullptr, Wx_w, Wx_b, G);
  gsn_cand_wmma<false><<<grd_mm, blk, 0, stream>>>(h1, h0, nullptr, W_w, W_b, G, CT);
  gsn_finalize<<<grd_fin, blk, 0, stream>>>(h1, CT, out + 1 * (size_t)TD);

  // ---- node 2: mailbox [h1, h0] (deg 2) ----
  gsn_gate_wmma<true><<<grd_mm, blk, 0, stream>>>(h2, h1, h0, Wx_w, Wx_b, G);
  gsn_cand_wmma<true><<<grd_mm, blk, 0, stream>>>(h2, h1, h0, W_w, W_b, G, CT);
  gsn_finalize<<<grd_fin, blk, 0, stream>>>(h2, CT, out + 2 * (size_t)TD);
}